// GAT_35794257445082
// MI455X (gfx1250) — compile-verified
//
#include <hip/hip_runtime.h>
#include <hip/hip_bf16.h>

typedef float v2f __attribute__((ext_vector_type(2)));
typedef float v8f __attribute__((ext_vector_type(8)));
typedef unsigned int v4u __attribute__((ext_vector_type(4)));
typedef int v4i __attribute__((ext_vector_type(4)));
typedef int v8i __attribute__((ext_vector_type(8)));

#define BATCH 4
#define NN    1024
#define DNODE 128
#define DEDGE 16
#define DGRAPH 128
#define OUTD  128
#define HEADS 8
#define DZ    256
#define WCOLS 272           // 128 (W_m) + 128 (W_skip) + 8 (W_a1) + 8 (W_a2)
#define JC    16            // j-chunk size in main kernel
#define NCH   (NN / JC)     // 64 chunks
#define EROW  260           // LDS floats per i-row of edge tile (256 + 4 pad)

#if __has_builtin(__builtin_amdgcn_tensor_load_to_lds) && \
    __has_builtin(__builtin_amdgcn_s_wait_tensorcnt)
  #define HAVE_TDM 1
  #if __has_include(<hip/amd_detail/amd_gfx1250_TDM.h>)
    #define TDM_6ARG 1
  #endif
#endif

// ---------------------------------------------------------------------------
// Kernel 1: fused prep GEMM  Z(4096x256) @ Wcat(256x272) -> OUTcat(4096x272)
// Z = [node_fts | hidden]; Wcat = [W_m | W_skip | W_a1 | W_a2] (virtual).
// One wave per 16x16 tile, V_WMMA_F32_16X16X4_F32, K split at the concat seam.
// ---------------------------------------------------------------------------
__global__ __launch_bounds__(32) void gat_prep_gemm(
    const float* __restrict__ node, const float* __restrict__ hidden,
    const float* __restrict__ Wm,  const float* __restrict__ bm,
    const float* __restrict__ Wsk, const float* __restrict__ bsk,
    const float* __restrict__ Wa1, const float* __restrict__ ba1,
    const float* __restrict__ Wa2, const float* __restrict__ ba2,
    float* __restrict__ outcat)
{
    int tile    = blockIdx.x;          // 0..(256*17-1)
    int rowT    = tile / 17;
    int colT    = tile % 17;
    int rowbase = rowT * 16;
    int lane    = threadIdx.x;
    int half    = lane >> 4;
    int l15     = lane & 15;
    int c       = colT * 16 + l15;     // output column 0..271

    const float* bsrc; int bld; float bv;
    if (c < 128)      { bsrc = Wm  + c;         bld = 128; bv = bm[c];        }
    else if (c < 256) { bsrc = Wsk + (c - 128); bld = 128; bv = bsk[c - 128]; }
    else if (c < 264) { bsrc = Wa1 + (c - 256); bld = 8;   bv = ba1[c - 256]; }
    else              { bsrc = Wa2 + (c - 264); bld = 8;   bv = ba2[c - 264]; }

    // lanes 0-15 supply WMMA K0/K1, lanes 16-31 K2/K3  ->  k0 = k + 2*half
    const float* ap = node + (size_t)(rowbase + l15) * DNODE + 2 * half;
    const float* bp = bsrc + (size_t)(2 * half) * bld;
    size_t bstep = (size_t)4 * bld;

    v8f acc = {};
#pragma unroll 4
    for (int k = 0; k < 128; k += 4) {             // node_fts half of z
        v2f a = *(const v2f*)ap;  ap += 4;
        v2f bb; bb.x = bp[0]; bb.y = bp[bld];  bp += bstep;
        acc = __builtin_amdgcn_wmma_f32_16x16x4_f32(false, a, false, bb,
                                                    (short)0, acc, false, false);
    }
    ap = hidden + (size_t)(rowbase + l15) * DNODE + 2 * half;
#pragma unroll 4
    for (int k = 0; k < 128; k += 4) {             // hidden half of z
        v2f a = *(const v2f*)ap;  ap += 4;
        v2f bb; bb.x = bp[0]; bb.y = bp[bld];  bp += bstep;
        acc = __builtin_amdgcn_wmma_f32_16x16x4_f32(false, a, false, bb,
                                                    (short)0, acc, false, false);
    }

    float* ocol = outcat + (size_t)rowbase * WCOLS + c;
#pragma unroll
    for (int r = 0; r < 8; ++r) {                  // C layout: M = r + 8*half, N = l15
        int m = r + 8 * half;
        ocol[(size_t)m * WCOLS] = acc[r] + bv;
    }
}

// ---------------------------------------------------------------------------
// Kernel 2: att_g[b][h] = b_ag[h] + graph_fts[b,:] @ W_ag[:,h]   (tiny)
// ---------------------------------------------------------------------------
__global__ __launch_bounds__(32) void gat_attg(
    const float* __restrict__ gf, const float* __restrict__ Wag,
    const float* __restrict__ bag, float* __restrict__ attg)
{
    int t = threadIdx.x;
    int b = t >> 3, h = t & 7;
    float s = bag[h];
    for (int k = 0; k < DGRAPH; ++k) s += gf[b * DGRAPH + k] * Wag[k * HEADS + h];
    attg[b * HEADS + h] = s;
}

// ---------------------------------------------------------------------------
// TDM: DMA one edge tile (16 i-rows x [16 j x 16 k] contiguous floats, row
// stride NN*16 elements) into LDS with 4-DWORD padding per 256-DWORD row so
// the WMMA A-operand ds_load_b64's are bank-conflict-free (row stride 260).
// ---------------------------------------------------------------------------
#if defined(HAVE_TDM)
__device__ __forceinline__ void tdm_load_edge_tile(const float* gsrc,
                                                   unsigned lds_off)
{
    unsigned long long ga = (unsigned long long)(size_t)gsrc;
    v4u g0;
    g0.x = 1u;                                        // count=1, user mode
    g0.y = lds_off;                                   // lds_addr (bytes)
    g0.z = (unsigned)(ga & 0xFFFFFFFFu);              // global_addr[31:0]
    g0.w = ((unsigned)(ga >> 32) & 0x01FFFFFFu)       // global_addr[56:32]
         | (2u << 30);                                // type = 2 ("image")
    v8i g1;
    g1[0] = (2 << 16)      // data_size = 4 bytes
          | (1 << 20)      // pad_enable
          | (7 << 22)      // pad_interval: 256 DWORDs
          | (3 << 25);     // pad_amount: 4 DWORDs
    g1[1] = (int)(256u << 16);   // tensor_dim0 = 256 (lo16 at bits 63:48)
    g1[2] = (int)(16u << 16);    // tensor_dim0 hi = 0 ; tensor_dim1 = 16
    g1[3] = (int)(256u << 16);   // tensor_dim1 hi = 0 ; tile_dim0 = 256
    g1[4] = 16;                  // tile_dim1 = 16 ; tile_dim2 = 0
    g1[5] = NN * DEDGE;          // tensor_dim0_stride = 16384 elements
    g1[6] = 0;
    g1[7] = 0;
    v4i z4 = {};
#if defined(TDM_6ARG)
    v8i z8 = {};
    __builtin_amdgcn_tensor_load_to_lds(g0, g1, z4, z4, z8, 0);
#else
    __builtin_amdgcn_tensor_load_to_lds(g0, g1, z4, z4, 0);
#endif
}
#endif

// ---------------------------------------------------------------------------
// Kernel 3: fused att_e + masked-exp softmax (no-max single pass) + coefs@values
//           + skip + relu. 256 blocks (b, 16-row i-tile) x 8 waves (one/head).
//           Edge tiles stream HBM -> LDS via double-buffered TDM.
// ---------------------------------------------------------------------------
__global__ __launch_bounds__(256) void gat_main(
    const float* __restrict__ edge, const float* __restrict__ adj,
    const float* __restrict__ Wae,  const float* __restrict__ bae,
    const float* __restrict__ outcat, const float* __restrict__ attg,
    float* __restrict__ out)
{
    __shared__ float s_edge[2][16 * EROW];  // [buf][i*260 + j*16 + k]
    __shared__ float s_attE[JC * 144];      // [j][i*9 + h]
    __shared__ float s_vals[JC * 136];      // [j][c], c < 128
    __shared__ float s_adj [JC * 17];       // [j][i]
    __shared__ float s_att2[JC * 8];        // [j][h]
    __shared__ float s_row [HEADS * 16];

    int blk   = blockIdx.x;                 // 0..255
    int b     = blk >> 6;
    int ibase = (blk & 63) << 4;
    int tid   = threadIdx.x;
    int h     = tid >> 5;                   // wave id == head
    int lane  = tid & 31;
    int half  = lane >> 4;
    int l15   = lane & 15;

    // B operand for att_e WMMA: W_ae padded to 16 cols (cols 8..15 = 0);
    // constant across all tiles -> registers.
    v2f wB[4];
#pragma unroll
    for (int ks = 0; ks < 4; ++ks) {
        int k0 = ks * 4 + 2 * half;
        if (l15 < 8) { wB[ks].x = Wae[k0 * HEADS + l15];
                       wB[ks].y = Wae[(k0 + 1) * HEADS + l15]; }
        else         { wB[ks].x = 0.f; wB[ks].y = 0.f; }
    }
    float biasE = (l15 < 8) ? bae[l15] : 0.f;

    float a1 = outcat[(size_t)(b * NN + ibase + l15) * WCOLS + 256 + h];
    float ag = attg[b * HEADS + h];

    const float* edgeB = edge + (size_t)b * NN * NN * DEDGE;
    const float* adjB  = adj  + (size_t)b * NN * NN;
    // start of this block's i-row band in edge (chunk c starts at +c*JC*DEDGE)
    const float* eband = edgeB + (size_t)ibase * NN * DEDGE;

#if defined(HAVE_TDM)
    unsigned lds_edge0 = (unsigned)(size_t)(void*)&s_edge[0][0];
    unsigned lds_edge1 = (unsigned)(size_t)(void*)&s_edge[1][0];
    if (h == 0)  // prologue: DMA chunk 0 into buffer 0
        tdm_load_edge_tile(eband, lds_edge0);
#endif

    v8f  acc = {};
    float rs = 0.f;

    for (int c = 0; c < NCH; ++c) {
        int jb = c * JC;
        // ---------------- stage small tiles into LDS -------------------
        {   // adjacency (transposed to [j][i])
            int i = tid >> 4, j = tid & 15;
            s_adj[j * 17 + i] = adjB[(size_t)(ibase + i) * NN + jb + j];
        }
        if (tid < 128) {   // att2 tile
            int j = tid >> 3, hh = tid & 7;
            s_att2[j * 8 + hh] =
                outcat[(size_t)(b * NN + jb + j) * WCOLS + 264 + hh];
        }
        {   // values tile (cols 0..127 of OUTcat), 2 float4 per thread
#pragma unroll
            for (int e = 0; e < 2; ++e) {
                int q = tid + 256 * e;            // 0..511
                int j = q >> 5, c0 = (q & 31) * 4;
                *(float4*)(s_vals + j * 136 + c0) =
                    *(const float4*)(outcat + (size_t)(b * NN + jb + j) * WCOLS + c0);
            }
        }
#if defined(HAVE_TDM)
        if (h == 0) {
            if (c + 1 < NCH)   // DMA next chunk into the other buffer
                tdm_load_edge_tile(eband + (size_t)(c + 1) * JC * DEDGE,
                                   ((c + 1) & 1) ? lds_edge1 : lds_edge0);
            if (c + 1 < NCH) __builtin_amdgcn_s_wait_tensorcnt((short)1);
            else             __builtin_amdgcn_s_wait_tensorcnt((short)0);
        }
#else
        {   // fallback: cooperative copy of this chunk's edge tile
            float* dst = &s_edge[c & 1][0];
#pragma unroll
            for (int s = 0; s < 4; ++s) {
                int q = tid + 256 * s;            // 0..1023 float4s
                int i = q >> 6, r4 = (q & 63) * 4;
                *(float4*)(dst + i * EROW + r4) =
                    *(const float4*)(eband + (size_t)i * NN * DEDGE
                                           + (size_t)jb * DEDGE + r4);
            }
        }
#endif
        __syncthreads();   // edge tile + small tiles visible to all waves

        // -------- phase A: att_e via f32 WMMA from LDS edge tile --------
        const float* eb = &s_edge[c & 1][0];
#pragma unroll
        for (int jj = 0; jj < 2; ++jj) {          // wave w handles j = 2w, 2w+1
            int j = h * 2 + jj;
            const float* er = eb + l15 * EROW + j * DEDGE;
            v8f eacc;
#pragma unroll
            for (int r = 0; r < 8; ++r) eacc[r] = biasE;
#pragma unroll
            for (int ks = 0; ks < 4; ++ks) {
                v2f a = *(const v2f*)(er + ks * 4 + 2 * half);
                eacc = __builtin_amdgcn_wmma_f32_16x16x4_f32(
                           false, a, false, wB[ks], (short)0, eacc, false, false);
            }
            if (l15 < 8) {
#pragma unroll
                for (int r = 0; r < 8; ++r) {
                    int i = r + 8 * half;
                    s_attE[j * 144 + i * 9 + l15] = eacc[r];
                }
            }
        }
        __syncthreads();

        // -------- phase B: masked exp + WMMA accumulate -----------------
#pragma unroll
        for (int jc = 0; jc < JC / 4; ++jc) {
            int j0 = jc * 4 + 2 * half;
            v2f p;
#pragma unroll
            for (int e = 0; e < 2; ++e) {
                int jl = j0 + e;
                float lg = a1 + s_attE[jl * 144 + l15 * 9 + h]
                              + s_att2[jl * 8 + h] + ag;
                lg = lg > 0.f ? lg : 0.01f * lg;              // leaky_relu
                float m  = s_adj[jl * 17 + l15];
                float pe = (m != 0.f) ? __expf(lg) : 0.f;     // -1e9 bias == mask
                p[e] = pe;
                rs  += pe;
            }
            v2f vb;
            vb.x = s_vals[j0 * 136 + h * 16 + l15];
            vb.y = s_vals[(j0 + 1) * 136 + h * 16 + l15];
            acc = __builtin_amdgcn_wmma_f32_16x16x4_f32(
                      false, p, false, vb, (short)0, acc, false, false);
        }
        __syncthreads();   // protect tiles before next chunk's staging
    }

    // row sums: lane L and L+16 hold partials of the same row (i = l15)
    rs += __shfl_xor(rs, 16, 32);
    if (half == 0) s_row[h * 16 + l15] = rs;
    __syncthreads();

    // epilogue: normalize, add skip, relu, store
#pragma unroll
    for (int r = 0; r < 8; ++r) {
        int m = r + 8 * half;
        float denom = s_row[h * 16 + m];
        float v = acc[r] / denom;
        size_t row = (size_t)(b * NN + ibase + m);
        v += outcat[row * WCOLS + 128 + h * 16 + l15];        // skip connection
        v = v > 0.f ? v : 0.f;
        out[row * OUTD + h * 16 + l15] = v;
    }
}

// ---------------------------------------------------------------------------
extern "C" void kernel_launch(void* const* d_in, const int* in_sizes, int n_in,
                              void* d_out, int out_size, void* d_ws, size_t ws_size,
                              hipStream_t stream)
{
    const float* node   = (const float*)d_in[0];
    const float* edge   = (const float*)d_in[1];
    const float* graph  = (const float*)d_in[2];
    const float* adj    = (const float*)d_in[3];
    const float* hidden = (const float*)d_in[4];
    const float* Wm     = (const float*)d_in[5];
    const float* bm     = (const float*)d_in[6];
    const float* Wsk    = (const float*)d_in[7];
    const float* bsk    = (const float*)d_in[8];
    const float* Wa1    = (const float*)d_in[9];
    const float* ba1    = (const float*)d_in[10];
    const float* Wa2    = (const float*)d_in[11];
    const float* ba2    = (const float*)d_in[12];
    const float* Wae    = (const float*)d_in[13];
    const float* bae    = (const float*)d_in[14];
    const float* Wag    = (const float*)d_in[15];
    const float* bag    = (const float*)d_in[16];

    float* ws     = (float*)d_ws;
    float* outcat = ws;                              // 4096 x 272 floats
    float* attg   = ws + (size_t)BATCH * NN * WCOLS; // 32 floats
    float* out    = (float*)d_out;

    gat_prep_gemm<<<256 * 17, 32, 0, stream>>>(node, hidden, Wm, bm, Wsk, bsk,
                                               Wa1, ba1, Wa2, ba2, outcat);
    gat_attg<<<1, 32, 0, stream>>>(graph, Wag, bag, attg);
    gat_main<<<BATCH * (NN / 16), 256, 0, stream>>>(edge, adj, Wae, bae,
                                                    outcat, attg, out);
}